// ScoreLoss_12687333392988
// MI455X (gfx1250) — compile-verified
//
#include <hip/hip_runtime.h>
#include <math.h>

#define BATCH 16
#define H 512
#define W 512
#define HW (H * W)
#define NPIX (BATCH * HW)
#define NCORN 500
#define NBLK 64                 /* H / 8 */
#define BLKPERIMG (NBLK * NBLK) /* 4096 */
#define CAND 4096
#define TPB 256
#define NTILES (BATCH * 32 * 32) /* 16384 workgroups of one 16x16 tile */
#define HSTRIDE 26               /* padded column stride for transposed h */

typedef float v2f __attribute__((ext_vector_type(2)));
typedef float v8f __attribute__((ext_vector_type(8)));

// Gaussian(7, sigma=1) weights as pure selects on immediates: no memory, no
// branches (v_cndmask).  With a compile-time index this folds to a constant.
__device__ __forceinline__ float g7w(int d) {
  float r = 0.0f;
  r = (d == 0 || d == 6) ? 0.0044330481f : r;
  r = (d == 1 || d == 5) ? 0.0540055826f : r;
  r = (d == 2 || d == 4) ? 0.2420362294f : r;
  r = (d == 3) ? 0.3990502797f : r;
  return r;
}

__device__ __forceinline__ int reflect512(int i) {
  i = (i < 0) ? -i : i;
  return (i >= 512) ? (1022 - i) : i;
}
__device__ __forceinline__ int clamp512(int i) {
  return i < 0 ? 0 : (i > 511 ? 511 : i);
}

// ---------------------------------------------------------------- K1: gray
__global__ __launch_bounds__(TPB) void gray_kernel(const float* __restrict__ imgs,
                                                   float* __restrict__ gray) {
  int gid = blockIdx.x * blockDim.x + threadIdx.x;
  if (gid >= NPIX) return;
  int b = gid / HW;
  int p = gid - b * HW;
  size_t ib = (size_t)b * 3 * HW + p;
  gray[gid] = 0.299f * imgs[ib] + 0.587f * imgs[ib + HW] + 0.114f * imgs[ib + 2 * HW];
}

// ------------------------------------------------- K2: sobel -> dx2,dy2,dxy
__global__ __launch_bounds__(TPB) void grad_kernel(const float* __restrict__ gray,
                                                   float* __restrict__ dx2,
                                                   float* __restrict__ dy2,
                                                   float* __restrict__ dxy) {
  int gid = blockIdx.x * blockDim.x + threadIdx.x;
  if (gid >= NPIX) return;
  int b = gid / HW;
  int p = gid - b * HW;
  int y = p / W, x = p - y * W;
  size_t base = (size_t)b * HW;
  int ym = clamp512(y - 1), yp = clamp512(y + 1);
  int xm = clamp512(x - 1), xp = clamp512(x + 1);
  float g00 = gray[base + ym * W + xm], g01 = gray[base + ym * W + x],
        g02 = gray[base + ym * W + xp];
  float g10 = gray[base + y * W + xm], g12 = gray[base + y * W + xp];
  float g20 = gray[base + yp * W + xm], g21 = gray[base + yp * W + x],
        g22 = gray[base + yp * W + xp];
  float dx = (-g00 + g02 - 2.f * g10 + 2.f * g12 - g20 + g22) * 0.125f;
  float dy = (-g00 - 2.f * g01 - g02 + g20 + 2.f * g21 + g22) * 0.125f;
  dx2[gid] = dx * dx;
  dy2[gid] = dy * dy;
  dxy[gid] = dx * dy;
}

// ---- helper: banded A fragments (Gm[m][k] = g7[k-m]), hoisted per lane ----
__device__ __forceinline__ void make_areg(v2f* Areg, int lane) {
  int laneHi = lane >> 4, lm = lane & 15;
#pragma unroll
  for (int kk = 0; kk < 6; ++kk) {
#pragma unroll
    for (int j = 0; j < 2; ++j) {
      Areg[kk][j] = g7w(kk * 4 + 2 * laneHi + j - lm);
    }
  }
}

// ---- helper: vertical 7-tap blur out(16x16) = Gm(16x24) * h(24x16) via ----
// chained V_WMMA_F32_16X16X4_F32.  h is stored transposed in LDS with
// column stride HSTRIDE (rows 22,23 zeroed) so each K-slice B fragment is
// one unconditional ds_load_b64.
__device__ __forceinline__ v8f blur_vert_wmma(const float* hT, const v2f* Areg,
                                              int lane) {
  int laneHi = lane >> 4, lm = lane & 15;
  v8f acc = {0.f, 0.f, 0.f, 0.f, 0.f, 0.f, 0.f, 0.f};
#pragma unroll
  for (int kk = 0; kk < 6; ++kk) {
    v2f Bv = *(const v2f*)(hT + lm * HSTRIDE + kk * 4 + 2 * laneHi);
    acc = __builtin_amdgcn_wmma_f32_16x16x4_f32(false, Areg[kk], false, Bv,
                                                (short)0, acc, false, false);
  }
  return acc;
}

// ---------------- K3: gauss7 blur of dx2/dy2/dxy (reflect) + GFTT score ----
__global__ __launch_bounds__(32) void blur_score_kernel(const float* __restrict__ dx2,
                                                        const float* __restrict__ dy2,
                                                        const float* __restrict__ dxy,
                                                        float* __restrict__ score) {
  __shared__ float in0[22 * 22], in1[22 * 22], in2[22 * 22];
  __shared__ __align__(16) float hT0[16 * HSTRIDE];
  __shared__ __align__(16) float hT1[16 * HSTRIDE];
  __shared__ __align__(16) float hT2[16 * HSTRIDE];
  int tx0 = blockIdx.x * 16, ty0 = blockIdx.y * 16, b = blockIdx.z;
  int lane = threadIdx.x;
  size_t ib = (size_t)b * HW;
  for (int t = lane; t < 22 * 22; t += 32) {
    int r = t / 22, c = t - r * 22;
    size_t off = ib + reflect512(ty0 - 3 + r) * W + reflect512(tx0 - 3 + c);
    in0[t] = dx2[off];
    in1[t] = dy2[off];
    in2[t] = dxy[off];
  }
  __syncthreads();
  // horizontal 7-tap; write transposed (column-major, stride HSTRIDE)
  for (int t = lane; t < 22 * 16; t += 32) {
    int r = t >> 4, c = t & 15;
    float s0 = 0.f, s1 = 0.f, s2 = 0.f;
#pragma unroll
    for (int k = 0; k < 7; ++k) {
      float w = g7w(k);
      int o = r * 22 + c + k;
      s0 += w * in0[o];
      s1 += w * in1[o];
      s2 += w * in2[o];
    }
    int o = c * HSTRIDE + r;
    hT0[o] = s0;
    hT1[o] = s1;
    hT2[o] = s2;
  }
  {  // zero K-padding rows 22,23 (exactly 32 slots -> one pass of the wave)
    int c = lane & 15, r = 22 + (lane >> 4);
    int o = c * HSTRIDE + r;
    hT0[o] = 0.f;
    hT1[o] = 0.f;
    hT2[o] = 0.f;
  }
  __syncthreads();
  v2f Areg[6];
  make_areg(Areg, lane);
  v8f a0 = blur_vert_wmma(hT0, Areg, lane);
  v8f a1 = blur_vert_wmma(hT1, Areg, lane);
  v8f a2 = blur_vert_wmma(hT2, Areg, lane);
  int laneHi = lane >> 4, lm = lane & 15;
#pragma unroll
  for (int r = 0; r < 8; ++r) {
    float p0 = a0[r], p1 = a1[r], p2 = a2[r];
    float tr = p0 + p1;
    float det = p0 * p1 - p2 * p2;
    float sc = 0.5f * (tr - sqrtf(fabsf(tr * tr - 4.0f * det)));
    score[ib + (ty0 + r + 8 * laneHi) * W + tx0 + lm] = sc;
  }
}

// ------------------------------------------------ K4: 5x5 NMS on score ----
__global__ __launch_bounds__(TPB) void nms_kernel(const float* __restrict__ score,
                                                  float* __restrict__ corners) {
  int gid = blockIdx.x * blockDim.x + threadIdx.x;
  if (gid >= NPIX) return;
  int b = gid / HW;
  int p = gid - b * HW;
  int y = p / W, x = p - y * W;
  size_t base = (size_t)b * HW;
  float v = score[gid];
  float m = -3.402823466e38f;
  for (int dy = -2; dy <= 2; ++dy) {
    int yy = y + dy;
    if ((unsigned)yy >= (unsigned)H) continue;
    for (int dx = -2; dx <= 2; ++dx) {
      int xx = x + dx;
      if ((unsigned)xx >= (unsigned)W) continue;
      m = fmaxf(m, score[base + yy * W + xx]);
    }
  }
  corners[gid] = (v == m) ? v : 0.0f;
}

// --------------------- K5: per-8x8-block max -> compacted candidate list ---
__global__ __launch_bounds__(TPB) void blockmax_kernel(const float* __restrict__ corners,
                                                       float* __restrict__ cval,
                                                       int* __restrict__ cidx,
                                                       int* __restrict__ cnt) {
  int gid = blockIdx.x * blockDim.x + threadIdx.x;
  if (gid >= BATCH * BLKPERIMG) return;
  int b = gid / BLKPERIMG;
  int blk = gid - b * BLKPERIMG;
  int by = blk >> 6, bx = blk & 63;
  size_t base = (size_t)b * HW;
  float m = 0.0f;
  int mi = -1;
  for (int r = 0; r < 8; ++r) {
    int y = by * 8 + r;
    for (int c = 0; c < 8; ++c) {
      int x = bx * 8 + c;
      float v = corners[base + y * W + x];
      if (v > m) {  // strict > keeps earliest index on (improbable) ties
        m = v;
        mi = y * W + x;
      }
    }
  }
  if (m > 0.0f) {
    int pos = atomicAdd(&cnt[b], 1);
    cval[b * CAND + pos] = m;
    cidx[b * CAND + pos] = mi;
  }
}

// ------------- K6: bitonic sort (value desc, index asc) + scatter top-500 --
__global__ __launch_bounds__(1024) void topk_scatter_kernel(const float* __restrict__ cval,
                                                            const int* __restrict__ cidx,
                                                            const int* __restrict__ cnt,
                                                            float* __restrict__ target) {
  __shared__ float sv[CAND];
  __shared__ int si[CAND];
  int b = blockIdx.x, tid = threadIdx.x;
  int n = cnt[b];
  for (int t = tid; t < CAND; t += 1024) {
    if (t < n) {
      sv[t] = cval[b * CAND + t];
      si[t] = cidx[b * CAND + t];
    } else {
      sv[t] = -1.0f;
      si[t] = 0x7fffffff;
    }
  }
  __syncthreads();
  for (int k = 2; k <= CAND; k <<= 1) {
    for (int j = k >> 1; j > 0; j >>= 1) {
      for (int i = tid; i < CAND; i += 1024) {
        int l = i ^ j;
        if (l > i) {
          float v1 = sv[i], v2 = sv[l];
          int i1 = si[i], i2 = si[l];
          bool before = (v1 > v2) || (v1 == v2 && i1 < i2);
          bool dir = ((i & k) == 0);
          if (before != dir) {
            sv[i] = v2;
            sv[l] = v1;
            si[i] = i2;
            si[l] = i1;
          }
        }
      }
      __syncthreads();
    }
  }
  if (tid < NCORN) {
    float v = sv[tid];
    if (v > 0.0f) target[(size_t)b * HW + si[tid]] = v;
  }
}

// ------------------------------- K7: 5x5 NMS on sparse target -> binary ----
__global__ __launch_bounds__(TPB) void nms_bin_kernel(const float* __restrict__ target,
                                                      float* __restrict__ binary) {
  int gid = blockIdx.x * blockDim.x + threadIdx.x;
  if (gid >= NPIX) return;
  int b = gid / HW;
  int p = gid - b * HW;
  int y = p / W, x = p - y * W;
  float v = target[gid];
  float o = 0.0f;
  if (v > 0.0f) {
    size_t base = (size_t)b * HW;
    float m = v;
    for (int dy = -2; dy <= 2; ++dy) {
      int yy = y + dy;
      if ((unsigned)yy >= (unsigned)H) continue;
      for (int dx = -2; dx <= 2; ++dx) {
        int xx = x + dx;
        if ((unsigned)xx >= (unsigned)W) continue;
        m = fmaxf(m, target[base + yy * W + xx]);
      }
    }
    o = (v == m) ? 1.0f : 0.0f;
  }
  binary[gid] = o;
}

// ---- K8: gauss7 blur of binary (WMMA) + fused BCE + laplacian partials ----
__global__ __launch_bounds__(32) void corner_loss_kernel(const float* __restrict__ binary,
                                                         const float* __restrict__ scores,
                                                         float* __restrict__ pb,
                                                         float* __restrict__ pt) {
  __shared__ float binl[22 * 22];
  __shared__ __align__(16) float hT[16 * HSTRIDE];
  __shared__ float sl[20 * 20];
  int tx0 = blockIdx.x * 16, ty0 = blockIdx.y * 16, b = blockIdx.z;
  int lane = threadIdx.x;
  size_t ib = (size_t)b * HW;
  for (int t = lane; t < 22 * 22; t += 32) {
    int r = t / 22, c = t - r * 22;
    binl[t] = binary[ib + reflect512(ty0 - 3 + r) * W + reflect512(tx0 - 3 + c)];
  }
  for (int t = lane; t < 20 * 20; t += 32) {
    int r = t / 20, c = t - r * 20;
    sl[t] = scores[ib + reflect512(ty0 - 2 + r) * W + reflect512(tx0 - 2 + c)];
  }
  __syncthreads();
  for (int t = lane; t < 22 * 16; t += 32) {
    int r = t >> 4, c = t & 15;
    float s = 0.f;
#pragma unroll
    for (int k = 0; k < 7; ++k) s += g7w(k) * binl[r * 22 + c + k];
    hT[c * HSTRIDE + r] = s;
  }
  {
    int c = lane & 15, r = 22 + (lane >> 4);
    hT[c * HSTRIDE + r] = 0.f;
  }
  __syncthreads();
  v2f Areg[6];
  make_areg(Areg, lane);
  v8f acc = blur_vert_wmma(hT, Areg, lane);
  int laneHi = lane >> 4, lm = lane & 15;
  float sb = 0.0f, st = 0.0f;
#pragma unroll
  for (int r = 0; r < 8; ++r) {
    int Mrow = r + 8 * laneHi;
    float c = acc[r];
    float s = sl[(Mrow + 2) * 20 + lm + 2];
    float S = 0.0f;
#pragma unroll
    for (int dy = -2; dy <= 2; ++dy)
#pragma unroll
      for (int dx = -2; dx <= 2; ++dx) S += sl[(Mrow + 2 + dy) * 20 + lm + 2 + dx];
    float lap = (S - 25.0f * s) * (1.0f / 48.0f);
    st += s * expf(-lap);
    float logp = fmaxf(logf(s), -100.0f);
    float l1mp = fmaxf(log1pf(-s), -100.0f);
    sb -= (c * logp + (1.0f - c) * l1mp);
  }
#pragma unroll
  for (int m = 16; m >= 1; m >>= 1) {
    sb += __shfl_xor(sb, m, 32);
    st += __shfl_xor(st, m, 32);
  }
  if (lane == 0) {
    int wg = ((int)blockIdx.z * gridDim.y + blockIdx.y) * gridDim.x + blockIdx.x;
    pb[wg] = sb;
    pt[wg] = st;
  }
}

// ---------------------------------- K9: deterministic final reduction ------
__global__ __launch_bounds__(256) void final_reduce_kernel(const float* __restrict__ pb,
                                                           const float* __restrict__ pt,
                                                           float* __restrict__ out) {
  __shared__ float s1[256], s2[256];
  float a = 0.f, c = 0.f;
  for (int i = threadIdx.x; i < NTILES; i += 256) {
    a += pb[i];
    c += pt[i];
  }
  s1[threadIdx.x] = a;
  s2[threadIdx.x] = c;
  __syncthreads();
  for (int m = 128; m >= 1; m >>= 1) {
    if ((int)threadIdx.x < m) {
      s1[threadIdx.x] += s1[threadIdx.x + m];
      s2[threadIdx.x] += s2[threadIdx.x + m];
    }
    __syncthreads();
  }
  if (threadIdx.x == 0) {
    float inv = 1.0f / (float)NPIX;
    out[0] = s1[0] * inv + 10.0f * (s2[0] * inv);
  }
}

extern "C" void kernel_launch(void* const* d_in, const int* in_sizes, int n_in,
                              void* d_out, int out_size, void* d_ws, size_t ws_size,
                              hipStream_t stream) {
  const float* scores = (const float*)d_in[0]; // (16,1,512,512)
  const float* imgs = (const float*)d_in[1];   // (16,3,512,512)
  float* ws = (float*)d_ws;
  const size_t M = (size_t)NPIX;
  // Buffer lifetimes (64 MB total):
  float* buf0 = ws;          // gray  -> score
  float* buf1 = ws + M;      // dx2   -> corners
  float* buf2 = ws + 2 * M;  // dy2   -> target -> partials
  float* buf3 = ws + 3 * M;  // dxy   -> candidates -> binary
  float* cval = buf3;
  int* cidx = (int*)(buf3 + BATCH * CAND);
  int* cnt = (int*)(buf3 + 2 * (size_t)BATCH * CAND);
  float* pb = buf2;
  float* pt = buf2 + NTILES;

  int nb = (NPIX + TPB - 1) / TPB;
  dim3 tgrid(W / 16, H / 16, BATCH);

  gray_kernel<<<nb, TPB, 0, stream>>>(imgs, buf0);
  grad_kernel<<<nb, TPB, 0, stream>>>(buf0, buf1, buf2, buf3);
  blur_score_kernel<<<tgrid, 32, 0, stream>>>(buf1, buf2, buf3, buf0);
  nms_kernel<<<nb, TPB, 0, stream>>>(buf0, buf1);
  (void)hipMemsetAsync(cnt, 0, BATCH * sizeof(int), stream);
  blockmax_kernel<<<(BATCH * BLKPERIMG + TPB - 1) / TPB, TPB, 0, stream>>>(buf1, cval, cidx, cnt);
  (void)hipMemsetAsync(buf2, 0, M * sizeof(float), stream); // zero target
  topk_scatter_kernel<<<BATCH, 1024, 0, stream>>>(cval, cidx, cnt, buf2);
  nms_bin_kernel<<<nb, TPB, 0, stream>>>(buf2, buf3); // binary overwrites cands
  corner_loss_kernel<<<tgrid, 32, 0, stream>>>(buf3, scores, pb, pt);
  final_reduce_kernel<<<1, 256, 0, stream>>>(pb, pt, (float*)d_out);
}